// WaveHaxGenerator1D_8048768713418
// MI455X (gfx1250) — compile-verified
//
#include <hip/hip_runtime.h>
#include <math.h>

// ---------------- problem constants (match reference) ----------------
#define SRATE   48000
#define HOP     480
#define NFFT    1920
#define NBINS   961          // NFFT/2 + 1
#define BB      2
#define FRAMES  200
#define TTOT    96000        // FRAMES*HOP
#define INC     192
#define GIN     256
#define NL      8
#define MAXH    120
#define CCH     NBINS        // 961
#define CCH2    (2*NBINS)    // 1922
#define PADL    720          // (NFFT-HOP)/2
#define ROWT    4            // 16-row tiles per wave in the WMMA matmul

#define CEILDIV(a,b) (((a)+(b)-1)/(b))

typedef float v2f __attribute__((ext_vector_type(2)));
typedef float v8f __attribute__((ext_vector_type(8)));

__device__ __forceinline__ float hannw(int n) {
    return 0.5f - 0.5f * cosf((float)n * (float)(6.283185307179586 / (double)NFFT));
}

// ---------------- 1. phase prefix per frame (f64, tiny serial) ----------------
__global__ void k_phase(const float* __restrict__ f0, const float* __restrict__ phase0,
                        double* __restrict__ pb) {
    int b = threadIdx.x;
    if (b >= BB) return;
    double acc = (double)phase0[0];
    for (int fr = 0; fr < FRAMES; ++fr) {
        pb[b * FRAMES + fr] = acc;   // phase0 + sum of full previous frames
        acc += (double)f0[b * FRAMES + fr] * (double)HOP / (double)SRATE;
    }
}

// ---------------- 2. PCPH prior waveform ----------------
__global__ void k_prior(const float* __restrict__ f0, const float* __restrict__ noise,
                        const double* __restrict__ pb, float* __restrict__ prior) {
    int i = blockIdx.x * blockDim.x + threadIdx.x;
    if (i >= BB * TTOT) return;
    int b = i / TTOT, t = i % TTOT;
    int fr = t / HOP, o = t % HOP;
    float f0v = f0[b * FRAMES + fr];
    float out = noise[i];
    if (f0v > 0.f) {
        double ph = pb[b * FRAMES + fr] + (double)(o + 1) * (double)f0v / (double)SRATE;
        double w  = 6.283185307179586476925286766559 * ph;
        double s  = 0.0;
        for (int k = 1; k <= MAXH; ++k) {
            if (f0v * (float)k <= 24000.f) s += sin(w * (double)k);
        }
        float nh  = 24000.f / f0v;
        float amp = 0.1f * sqrtf(2.f / nh);
        out += amp * (float)s;
    }
    prior[i] = out;
}

// ---------------- 3. DFT twiddle tables (exact integer angle reduction) ----------------
__global__ void k_dfttab(float* __restrict__ cosT, float* __restrict__ sinT) {
    int i = blockIdx.x * blockDim.x + threadIdx.x;
    if (i >= NBINS * NFFT) return;
    int k = i / NFFT, n = i % NFFT;
    int m = (int)(((long long)k * (long long)n) % (long long)NFFT);
    float th = (float)m * (float)(6.283185307179586 / (double)NFFT);
    cosT[i] = cosf(th);
    sinT[i] = sinf(th);
}

// ---------------- 4. frame + window:  fv[b][n][fr] ----------------
__global__ void k_frame(const float* __restrict__ prior, float* __restrict__ fv) {
    int i = blockIdx.x * blockDim.x + threadIdx.x;
    if (i >= BB * NFFT * FRAMES) return;
    int b = i / (NFFT * FRAMES);
    int r = i % (NFFT * FRAMES);
    int n = r / FRAMES, fr = r % FRAMES;
    int s = HOP * fr + n - PADL;
    float v = (s >= 0 && s < TTOT) ? prior[b * TTOT + s] : 0.f;
    fv[i] = v * hannw(n);
}

// ---------------- 5. generic FP32 WMMA matmul (TRANSA templated) ----------------
// D[b][m][n] (=|+=) alpha * sum_k A[m][k]*Bm[b][k][n]  (+bias[m]) (gelu)
// TRANSA: A[m][k] is read at A[k*NFFT + m]  (iDFT with the cos/sin tables)
//
// Each wave owns ROWT consecutive 16-row tiles x one 16-col tile.
// Main K loop is branchless with carried pointers + immediate offsets; OOB
// rows/cols use clamped addresses (garbage lands only in never-stored outputs).
// Only the K%4 tail needs value masking.
template<int TRANSA>
__global__ void __launch_bounds__(256)
k_mm(const float* __restrict__ A, const float* __restrict__ Bm,
     float* __restrict__ D, const float* __restrict__ bias,
     int O, int K, long long strideB, float alpha, int gelu, int accum) {
    constexpr int F = FRAMES;            // all call sites use 200 columns
    constexpr int LDA = NFFT;            // transA leading dim (cos/sin tables)
    int wave = threadIdx.x >> 5;
    int lane = threadIdx.x & 31;
    int half = lane >> 4;
    int idx  = lane & 15;
    int base = (blockIdx.x * (blockDim.x >> 5) + wave) * (16 * ROWT);
    if (base >= O) return;               // whole-wave exit: EXEC uniform for WMMA
    int c0 = blockIdx.y * 16;
    int b  = blockIdx.z;
    const float* Bp = Bm + (long long)b * strideB;
    int col  = c0 + idx;
    int colc = (col < F) ? col : (F - 1);        // clamped address for OOB lanes

    int rowc[ROWT];
#pragma unroll
    for (int t = 0; t < ROWT; ++t) {
        int r = base + t * 16 + idx;
        rowc[t] = (r < O) ? r : (O - 1);         // clamped address for OOB rows
    }

    v8f acc[ROWT];
#pragma unroll
    for (int t = 0; t < ROWT; ++t) acc[t] = (v8f){0.f,0.f,0.f,0.f,0.f,0.f,0.f,0.f};

    // carried pointers: per-iteration address math is just 5 u64 adds
    int kh = half * 2;                   // this lane's first K slot within each group of 4
    const float* bptr = Bp + (long long)kh * F + colc;
    const float* aptr[ROWT];
#pragma unroll
    for (int t = 0; t < ROWT; ++t)
        aptr[t] = TRANSA ? (A + (long long)kh * LDA + rowc[t])
                         : (A + (long long)rowc[t] * K + kh);

    int K4 = K & ~3;
#pragma unroll 2
    for (int k0 = 0; k0 < K4; k0 += 4) {
        // prefetch the weight stream ~32 K ahead (global_prefetch_b8, speculative)
        __builtin_prefetch(TRANSA ? (aptr[0] + 32 * (long long)LDA) : (aptr[0] + 32), 0, 1);
        float b0 = bptr[0];
        float b1 = bptr[F];
        v2f bv; bv[0] = b0; bv[1] = b1;
#pragma unroll
        for (int t = 0; t < ROWT; ++t) {
            float a0 = aptr[t][0];
            float a1 = TRANSA ? aptr[t][LDA] : aptr[t][1];
            v2f av; av[0] = a0; av[1] = a1;
            acc[t] = __builtin_amdgcn_wmma_f32_16x16x4_f32(false, av, false, bv,
                                                           (short)0, acc[t], false, false);
            aptr[t] += TRANSA ? (long long)4 * LDA : 4;
        }
        bptr += (long long)4 * F;
    }
    if (K4 < K) {                                // tail group: value-masked, branchless
        int ka  = K4 + kh;
        int ka0 = (ka     < K) ? ka     : (K - 1);
        int ka1 = (ka + 1 < K) ? ka + 1 : (K - 1);
        float b0 = Bp[(long long)ka0 * F + colc];
        float b1 = Bp[(long long)ka1 * F + colc];
        b0 = (ka     < K) ? b0 : 0.f;
        b1 = (ka + 1 < K) ? b1 : 0.f;
        v2f bv; bv[0] = b0; bv[1] = b1;
#pragma unroll
        for (int t = 0; t < ROWT; ++t) {
            float a0, a1;
            if (TRANSA) {
                a0 = A[(long long)ka0 * LDA + rowc[t]];
                a1 = A[(long long)ka1 * LDA + rowc[t]];
            } else {
                a0 = A[(long long)rowc[t] * K + ka0];
                a1 = A[(long long)rowc[t] * K + ka1];
            }
            a0 = (ka     < K) ? a0 : 0.f;
            a1 = (ka + 1 < K) ? a1 : 0.f;
            v2f av; av[0] = a0; av[1] = a1;
            acc[t] = __builtin_amdgcn_wmma_f32_16x16x4_f32(false, av, false, bv,
                                                           (short)0, acc[t], false, false);
        }
    }

    if (col >= F) return;
    float* Dp = D + (long long)b * (long long)O * F;
#pragma unroll
    for (int t = 0; t < ROWT; ++t) {
#pragma unroll
        for (int r = 0; r < 8; ++r) {
            int m = base + t * 16 + r + half * 8;
            if (m < O) {
                float v = alpha * acc[t][r];
                if (bias) v += bias[m];
                if (gelu) v = 0.5f * v * (1.f + erff(v * 0.70710678118654752440f));
                long long di = (long long)m * F + col;
                if (accum) Dp[di] += v; else Dp[di] = v;
            }
        }
    }
}

// ---------------- 6. im2col with reflect padding (pad=3, ksize=7) ----------------
__global__ void k_im2col(const float* __restrict__ X, float* __restrict__ colb, int Cin) {
    int tot = BB * Cin * 7 * FRAMES;
    int i = blockIdx.x * blockDim.x + threadIdx.x;
    if (i >= tot) return;
    int b  = i / (Cin * 7 * FRAMES);
    int r  = i % (Cin * 7 * FRAMES);
    int cc = r / FRAMES, f = r % FRAMES;
    int ci = cc / 7, j = cc % 7;
    int s = f + j - 3;
    if (s < 0) s = -s;
    if (s >= FRAMES) s = 2 * FRAMES - 2 - s;
    colb[i] = X[((long long)b * Cin + ci) * FRAMES + s];
}

// ---------------- 7. conditioning vector + broadcast add ----------------
__global__ void k_cond(const float* __restrict__ cw, const float* __restrict__ cb,
                       const float* __restrict__ g, float* __restrict__ cv) {
    int i = blockIdx.x * blockDim.x + threadIdx.x;
    if (i >= BB * CCH) return;
    int b = i / CCH, c = i % CCH;
    float s = cb[c];
    for (int k = 0; k < GIN; ++k) s += cw[c * GIN + k] * g[b * GIN + k];
    cv[i] = s;
}

__global__ void k_addbc(float* __restrict__ h, const float* __restrict__ cv) {
    int i = blockIdx.x * blockDim.x + threadIdx.x;
    if (i >= BB * CCH * FRAMES) return;
    int b = i / (CCH * FRAMES);
    int c = (i / FRAMES) % CCH;
    h[i] += cv[b * CCH + c];
}

// ---------------- 8. concat [p1,p2,q1,q2,h] over channels ----------------
__global__ void k_concat(const float* __restrict__ p1, const float* __restrict__ p2,
                         const float* __restrict__ q1, const float* __restrict__ q2,
                         const float* __restrict__ h,  float* __restrict__ cat) {
    int tot = BB * 5 * CCH * FRAMES;
    int i = blockIdx.x * blockDim.x + threadIdx.x;
    if (i >= tot) return;
    int b  = i / (5 * CCH * FRAMES);
    int r  = i % (5 * CCH * FRAMES);
    int cc = r / FRAMES, f = r % FRAMES;
    int grp = cc / CCH, c = cc % CCH;
    const float* src = (grp == 0) ? p1 : (grp == 1) ? p2 : (grp == 2) ? q1 : (grp == 3) ? q2 : h;
    cat[i] = src[((long long)b * CCH + c) * FRAMES + f];
}

// ---------------- 9. depthwise conv7, zero pad ----------------
__global__ void k_dwconv(const float* __restrict__ h, const float* __restrict__ w7,
                         const float* __restrict__ bias, float* __restrict__ out) {
    int i = blockIdx.x * blockDim.x + threadIdx.x;
    if (i >= BB * CCH * FRAMES) return;
    int b = i / (CCH * FRAMES);
    int r = i % (CCH * FRAMES);
    int c = r / FRAMES, f = r % FRAMES;
    float s = bias[c];
    const float* hp = h + ((long long)b * CCH + c) * FRAMES;
#pragma unroll
    for (int j = 0; j < 7; ++j) {
        int sf = f + j - 3;
        if (sf >= 0 && sf < FRAMES) s += w7[c * 7 + j] * hp[sf];
    }
    out[i] = s;
}

// ---------------- 10. LayerNorm over channels (two-pass, in place) ----------------
__global__ void __launch_bounds__(256)
k_ln(float* __restrict__ d, const float* __restrict__ gam, const float* __restrict__ bet) {
    __shared__ float red[256];
    int bf = blockIdx.x;
    int b = bf / FRAMES, f = bf % FRAMES;
    float* base = d + (long long)b * CCH * FRAMES + f;
    float s = 0.f;
    for (int c = threadIdx.x; c < CCH; c += blockDim.x) s += base[(long long)c * FRAMES];
    red[threadIdx.x] = s; __syncthreads();
    for (int o = 128; o > 0; o >>= 1) { if (threadIdx.x < o) red[threadIdx.x] += red[threadIdx.x + o]; __syncthreads(); }
    float mu = red[0] / (float)CCH;
    __syncthreads();
    s = 0.f;
    for (int c = threadIdx.x; c < CCH; c += blockDim.x) { float x = base[(long long)c * FRAMES] - mu; s += x * x; }
    red[threadIdx.x] = s; __syncthreads();
    for (int o = 128; o > 0; o >>= 1) { if (threadIdx.x < o) red[threadIdx.x] += red[threadIdx.x + o]; __syncthreads(); }
    float inv = rsqrtf(red[0] / (float)CCH + 1e-5f);
    for (int c = threadIdx.x; c < CCH; c += blockDim.x) {
        float x = base[(long long)c * FRAMES];
        base[(long long)c * FRAMES] = (x - mu) * inv * gam[c] + bet[c];
    }
}

// ---------------- 11. GRN ----------------
__global__ void k_grn_gx(const float* __restrict__ t2, float* __restrict__ gx) {
    int i = blockIdx.x * blockDim.x + threadIdx.x;
    if (i >= BB * CCH2) return;
    const float* p = t2 + (long long)i * FRAMES;
    float s = 0.f;
    for (int f = 0; f < FRAMES; ++f) { float v = p[f]; s += v * v; }
    gx[i] = sqrtf(s);
}

__global__ void __launch_bounds__(256)
k_grn_mu(const float* __restrict__ gx, float* __restrict__ gmu) {
    __shared__ float red[256];
    int b = blockIdx.x;
    float s = 0.f;
    for (int c = threadIdx.x; c < CCH2; c += blockDim.x) s += gx[b * CCH2 + c];
    red[threadIdx.x] = s; __syncthreads();
    for (int o = 128; o > 0; o >>= 1) { if (threadIdx.x < o) red[threadIdx.x] += red[threadIdx.x + o]; __syncthreads(); }
    if (threadIdx.x == 0) gmu[b] = red[0] / (float)CCH2;
}

__global__ void k_grn_apply(float* __restrict__ t2, const float* __restrict__ gx,
                            const float* __restrict__ gmu, const float* __restrict__ gg,
                            const float* __restrict__ gb) {
    int i = blockIdx.x * blockDim.x + threadIdx.x;
    if (i >= BB * CCH2 * FRAMES) return;
    int b = i / (CCH2 * FRAMES);
    int c = (i / FRAMES) % CCH2;
    float nx = gx[b * CCH2 + c] / (gmu[b] + 1e-6f);
    float v = t2[i];
    t2[i] = v * (1.f + gg[c] * nx) + gb[c];
}

// ---------------- 12. fold irfft Hermitian weights / sign / 1/N into spectrum ----------------
__global__ void k_prescale(float* __restrict__ o) {
    int i = blockIdx.x * blockDim.x + threadIdx.x;
    if (i >= BB * CCH2 * FRAMES) return;
    int r = i % (CCH2 * FRAMES);
    int k = r / FRAMES;
    int kk = (k < NBINS) ? k : k - NBINS;
    float wk = (kk == 0 || kk == NBINS - 1) ? 1.f : 2.f;
    float sc = wk * (1.f / (float)NFFT);
    if (k >= NBINS) sc = -sc;            // x[n] = (1/N) sum wk*(re*cos - im*sin)
    o[i] *= sc;
}

// ---------------- 13. windowed overlap-add + envelope normalize (gather, no atomics) ----------------
__global__ void k_ola(const float* __restrict__ xf, float* __restrict__ out) {
    int i = blockIdx.x * blockDim.x + threadIdx.x;
    if (i >= BB * TTOT) return;
    int b = i / TTOT, t = i % TTOT;
    int pos = t + PADL;
    int frhi = pos / HOP; if (frhi > FRAMES - 1) frhi = FRAMES - 1;
    int frlo = (pos - (NFFT - 1) + (HOP - 1)) / HOP; if (frlo < 0) frlo = 0;
    float num = 0.f, den = 0.f;
    for (int fr = frlo; fr <= frhi; ++fr) {
        int n = pos - HOP * fr;          // 0..NFFT-1
        float w = hannw(n);
        num += xf[((long long)b * NFFT + n) * FRAMES + fr] * w;
        den += w * w;
    }
    out[i] = num / den;
}

// ---------------- host orchestration ----------------
extern "C" void kernel_launch(void* const* d_in, const int* in_sizes, int n_in,
                              void* d_out, int out_size, void* d_ws, size_t ws_size,
                              hipStream_t stream) {
    (void)in_sizes; (void)n_in; (void)out_size; (void)ws_size;
    const float* x      = (const float*)d_in[0];
    const float* f0     = (const float*)d_in[1];
    const float* g      = (const float*)d_in[2];
    const float* noise  = (const float*)d_in[3];
    const float* phase0 = (const float*)d_in[4];
    const float* pp1_w  = (const float*)d_in[5];
    const float* pp1_b  = (const float*)d_in[6];
    const float* pp2_w  = (const float*)d_in[7];
    const float* pp2_b  = (const float*)d_in[8];
    const float* cp_w   = (const float*)d_in[9];
    const float* cp_b   = (const float*)d_in[10];
    const float* cond_w = (const float*)d_in[11];
    const float* cond_b = (const float*)d_in[12];
    const float* ip_w   = (const float*)d_in[13];
    const float* c1_w   = (const float*)d_in[14];
    const float* c1_b   = (const float*)d_in[15];
    const float* ln_g   = (const float*)d_in[16];
    const float* ln_b   = (const float*)d_in[17];
    const float* c2_w   = (const float*)d_in[18];
    const float* c2_b   = (const float*)d_in[19];
    const float* grn_g  = (const float*)d_in[20];
    const float* grn_b  = (const float*)d_in[21];
    const float* c3_w   = (const float*)d_in[22];
    const float* c3_b   = (const float*)d_in[23];
    const float* op_w   = (const float*)d_in[24];
    const float* op_b   = (const float*)d_in[25];
    float* out = (float*)d_out;

    // carve workspace (~56 MB)
    char* wp = (char*)d_ws;
    auto alloc = [&](size_t bytes) -> char* {
        char* r = wp; wp += (bytes + 255) & ~(size_t)255; return r;
    };
    double* pb   = (double*)alloc((size_t)BB * FRAMES * 8);
    float* prior = (float*)alloc((size_t)BB * TTOT * 4);
    float* cosT  = (float*)alloc((size_t)NBINS * NFFT * 4);
    float* sinT  = (float*)alloc((size_t)NBINS * NFFT * 4);
    float* fv    = (float*)alloc((size_t)BB * NFFT * FRAMES * 4);
    float* p1    = (float*)alloc((size_t)BB * CCH * FRAMES * 4);
    float* p2    = (float*)alloc((size_t)BB * CCH * FRAMES * 4);
    float* q1    = (float*)alloc((size_t)BB * CCH * FRAMES * 4);
    float* q2    = (float*)alloc((size_t)BB * CCH * FRAMES * 4);
    float* h     = (float*)alloc((size_t)BB * CCH * FRAMES * 4);
    float* dwout = (float*)alloc((size_t)BB * CCH * FRAMES * 4);
    float* colb  = (float*)alloc((size_t)BB * CCH * 7 * FRAMES * 4);
    float* cat   = (float*)alloc((size_t)BB * 5 * CCH * FRAMES * 4);
    float* cv    = (float*)alloc((size_t)BB * CCH * 4);
    float* t2    = (float*)alloc((size_t)BB * CCH2 * FRAMES * 4);
    float* gx    = (float*)alloc((size_t)BB * CCH2 * 4);
    float* gmu   = (float*)alloc((size_t)BB * 4);
    float* obuf  = (float*)alloc((size_t)BB * CCH2 * FRAMES * 4);
    float* xf    = (float*)alloc((size_t)BB * NFFT * FRAMES * 4);

    auto mm = [&](const float* A, const float* Bm, float* D, const float* bias,
                  int O, int K, int transA, long long strideB,
                  float alpha, int gelu, int accum) {
        dim3 grid(CEILDIV(O, 16 * ROWT * 8), CEILDIV(FRAMES, 16), BB);
        if (transA)
            k_mm<1><<<grid, 256, 0, stream>>>(A, Bm, D, bias, O, K, strideB, alpha, gelu, accum);
        else
            k_mm<0><<<grid, 256, 0, stream>>>(A, Bm, D, bias, O, K, strideB, alpha, gelu, accum);
    };

    // ---- prior + STFT ----
    k_phase <<<1, 32, 0, stream>>>(f0, phase0, pb);
    k_prior <<<CEILDIV(BB * TTOT, 256), 256, 0, stream>>>(f0, noise, pb, prior);
    k_dfttab<<<CEILDIV(NBINS * NFFT, 256), 256, 0, stream>>>(cosT, sinT);
    k_frame <<<CEILDIV(BB * NFFT * FRAMES, 256), 256, 0, stream>>>(prior, fv);
    mm(cosT, fv, p1, nullptr, NBINS, NFFT, 0, (long long)NFFT * FRAMES,  1.f, 0, 0);
    mm(sinT, fv, p2, nullptr, NBINS, NFFT, 0, (long long)NFFT * FRAMES, -1.f, 0, 0);

    // ---- prior/cond projections ----
    k_im2col<<<CEILDIV(BB * CCH * 7 * FRAMES, 256), 256, 0, stream>>>(p1, colb, CCH);
    mm(pp1_w, colb, q1, pp1_b, CCH, CCH * 7, 0, (long long)CCH * 7 * FRAMES, 1.f, 0, 0);
    k_im2col<<<CEILDIV(BB * CCH * 7 * FRAMES, 256), 256, 0, stream>>>(p2, colb, CCH);
    mm(pp2_w, colb, q2, pp2_b, CCH, CCH * 7, 0, (long long)CCH * 7 * FRAMES, 1.f, 0, 0);
    k_im2col<<<CEILDIV(BB * INC * 7 * FRAMES, 256), 256, 0, stream>>>(x, colb, INC);
    mm(cp_w, colb, h, cp_b, CCH, INC * 7, 0, (long long)INC * 7 * FRAMES, 1.f, 0, 0);
    k_cond <<<CEILDIV(BB * CCH, 256), 256, 0, stream>>>(cond_w, cond_b, g, cv);
    k_addbc<<<CEILDIV(BB * CCH * FRAMES, 256), 256, 0, stream>>>(h, cv);
    k_concat<<<CEILDIV(BB * 5 * CCH * FRAMES, 256), 256, 0, stream>>>(p1, p2, q1, q2, h, cat);
    mm(ip_w, cat, h, nullptr, CCH, 5 * CCH, 0, (long long)5 * CCH * FRAMES, 1.f, 0, 0);

    // ---- ConvNeXt blocks ----
    for (int i = 0; i < NL; ++i) {
        k_dwconv<<<CEILDIV(BB * CCH * FRAMES, 256), 256, 0, stream>>>(
            h, c1_w + (size_t)i * CCH * 7, c1_b + (size_t)i * CCH, dwout);
        k_ln<<<BB * FRAMES, 256, 0, stream>>>(dwout, ln_g + (size_t)i * CCH, ln_b + (size_t)i * CCH);
        mm(c2_w + (size_t)i * CCH2 * CCH, dwout, t2, c2_b + (size_t)i * CCH2,
           CCH2, CCH, 0, (long long)CCH * FRAMES, 1.f, 1, 0);
        k_grn_gx<<<CEILDIV(BB * CCH2, 256), 256, 0, stream>>>(t2, gx);
        k_grn_mu<<<BB, 256, 0, stream>>>(gx, gmu);
        k_grn_apply<<<CEILDIV(BB * CCH2 * FRAMES, 256), 256, 0, stream>>>(
            t2, gx, gmu, grn_g + (size_t)i * CCH2, grn_b + (size_t)i * CCH2);
        mm(c3_w + (size_t)i * CCH * CCH2, t2, h, c3_b + (size_t)i * CCH,
           CCH, CCH2, 0, (long long)CCH2 * FRAMES, 1.f, 0, 1);   // += residual
    }

    // ---- output proj + iSTFT ----
    mm(op_w, h, obuf, op_b, CCH2, CCH, 0, (long long)CCH * FRAMES, 1.f, 0, 0);
    k_prescale<<<CEILDIV(BB * CCH2 * FRAMES, 256), 256, 0, stream>>>(obuf);
    mm(cosT, obuf,                          xf, nullptr, NFFT, NBINS, 1,
       (long long)CCH2 * FRAMES, 1.f, 0, 0);
    mm(sinT, obuf + (size_t)NBINS * FRAMES, xf, nullptr, NFFT, NBINS, 1,
       (long long)CCH2 * FRAMES, 1.f, 0, 1);
    k_ola<<<CEILDIV(BB * TTOT, 256), 256, 0, stream>>>(xf, out);
}